// TopologicalAdversarialLoss_40364102648253
// MI455X (gfx1250) — compile-verified
//
#include <hip/hip_runtime.h>
#include <hip/hip_bf16.h>

typedef __attribute__((ext_vector_type(2))) float v2f;
typedef __attribute__((ext_vector_type(8))) float v8f;

#define NPIX   4096   // 64*64
#define NE     8064   // 2*64*64 - 64 - 64
#define NHORIZ 4032   // 64 rows * 63
#define NSORT  8192   // next pow2 >= NE

// One block per (image, source): blocks 0..15 -> real, 16..31 -> synthetic.
// Computes the 0-dim cubical persistence polynomial statistic for one image.
__global__ __launch_bounds__(256) void ph0_stat_kernel(const float* __restrict__ real,
                                                       const float* __restrict__ syn,
                                                       float* __restrict__ stats) {
  __shared__ float          vals[NPIX];     // 16 KB
  __shared__ unsigned int   skey[NSORT];    // 32 KB  (float bits of edge death value)
  __shared__ unsigned short sid[NSORT];     // 16 KB  (edge id)
  __shared__ int            parent[NPIX];   // 16 KB
  __shared__ float          contrib[NPIX];  // 16 KB  (per-merge stat terms)

  const int tid = threadIdx.x;
  const int b   = blockIdx.x;
  const float* src = (b < 16) ? (real + b * NPIX) : (syn + (b - 16) * NPIX);

  // CDNA5 prefetch of this block's tile (global_prefetch_b8).
  __builtin_prefetch(src + tid * 16, 0, 3);

  for (int i = tid; i < NPIX; i += 256) {
    vals[i]    = src[i];
    parent[i]  = i;
    contrib[i] = 0.0f;
  }
  __syncthreads();

  // Build (death_key, edge_id) pairs; pad to NSORT with +max keys.
  for (int i = tid; i < NSORT; i += 256) {
    if (i < NE) {
      int u, v;
      if (i < NHORIZ) { int r = i / 63; int c = i - r * 63; u = r * 64 + c; v = u + 1; }
      else            { u = i - NHORIZ; v = u + 64; }
      float w = fmaxf(vals[u], vals[v]);   // values >= 0 -> bit pattern order == numeric order
      skey[i] = __float_as_uint(w);
      sid[i]  = (unsigned short)i;
    } else {
      skey[i] = 0xFFFFFFFFu;
      sid[i]  = 0xFFFFu;
    }
  }

  // In-LDS bitonic sort, ascending by key.
  for (unsigned int k = 2; k <= NSORT; k <<= 1) {
    for (unsigned int j = k >> 1; j > 0; j >>= 1) {
      __syncthreads();
      for (unsigned int i = tid; i < NSORT; i += 256) {
        unsigned int ixj = i ^ j;
        if (ixj > i) {
          unsigned int ka = skey[i], kb = skey[ixj];
          bool up     = ((i & k) == 0);
          bool doswap = up ? (ka > kb) : (ka < kb);
          if (doswap) {
            skey[i] = kb; skey[ixj] = ka;
            unsigned short t = sid[i]; sid[i] = sid[ixj]; sid[ixj] = t;
          }
        }
      }
    }
  }
  __syncthreads();

  // Serial Kruskal union-find with elder rule (lane 0). Persistence pairs are
  // invariant to equal-key ordering, so any correct sort matches the reference.
  if (tid == 0) {
    int m = 0;
    for (int i = 0; i < NE; ++i) {
      int e = sid[i];
      int u, v;
      if (e < NHORIZ) { int r = e / 63; int c = e - r * 63; u = r * 64 + c; v = u + 1; }
      else            { u = e - NHORIZ; v = u + 64; }
      int ru = u; while (parent[ru] != ru) { parent[ru] = parent[parent[ru]]; ru = parent[ru]; }
      int rv = v; while (parent[rv] != rv) { parent[rv] = parent[parent[rv]]; rv = parent[rv]; }
      if (ru != rv) {
        float vu = vals[ru], vv = vals[rv];
        int elder = (vu <= vv) ? ru : rv;
        int young = (vu <= vv) ? rv : ru;
        parent[young] = elder;
        parent[u] = elder;
        parent[v] = elder;
        float bth  = vals[young];                 // birth = younger root's min pixel
        float dth  = __uint_as_float(skey[i]);    // death = edge weight
        float pers = dth - bth;
        float mid  = 0.5f * (dth + bth);
        contrib[m++] = (pers * pers) * (mid * mid);
      }
    }
  }
  __syncthreads();

  // Reduce 4096 contributions with V_WMMA_F32_16X16X4_F32 (wave 0, EXEC all-1s).
  // A = 16x4 tile of data, B = ones -> each D row holds its row-sum in all 16
  // columns; grand total = (sum of the 16x16 accumulator) / 16 (exact in f32).
  if (tid < 32) {
    v2f ones; ones.x = 1.0f; ones.y = 1.0f;
    v8f acc = {0.f, 0.f, 0.f, 0.f, 0.f, 0.f, 0.f, 0.f};
    for (int c = 0; c < NPIX / 64; ++c) {
      int base = c * 64 + tid * 2;
      v2f a; a.x = contrib[base]; a.y = contrib[base + 1];
      acc = __builtin_amdgcn_wmma_f32_16x16x4_f32(
          /*neg_a=*/false, a, /*neg_b=*/false, ones,
          /*c_mod=*/(short)0, acc, /*reuse_a=*/false, /*reuse_b=*/false);
    }
    float s = acc[0] + acc[1] + acc[2] + acc[3] + acc[4] + acc[5] + acc[6] + acc[7];
    for (int off = 16; off > 0; off >>= 1) s += __shfl_down(s, off, 32);
    if (tid == 0) stats[b] = s * 0.0625f;   // /16
  }
}

// loss = sum_b |stat_syn[b] - stat_real[b]|
__global__ __launch_bounds__(32) void summary_loss_kernel(const float* __restrict__ stats,
                                                          float* __restrict__ out) {
  int t = threadIdx.x;
  float d = 0.0f;
  if (t < 16) d = fabsf(stats[16 + t] - stats[t]);
  for (int off = 16; off > 0; off >>= 1) d += __shfl_down(d, off, 32);
  if (t == 0) out[0] = d;
}

extern "C" void kernel_launch(void* const* d_in, const int* in_sizes, int n_in,
                              void* d_out, int out_size, void* d_ws, size_t ws_size,
                              hipStream_t stream) {
  const float* real = (const float*)d_in[0];   // [16,1,64,64] f32
  const float* syn  = (const float*)d_in[1];   // [16,1,64,64] f32
  float* stats = (float*)d_ws;                 // 32 floats of scratch
  float* out   = (float*)d_out;                // scalar f32

  ph0_stat_kernel<<<32, 256, 0, stream>>>(real, syn, stats);
  summary_loss_kernel<<<1, 32, 0, stream>>>(stats, out);
}